// PScanTriton_37933151158406
// MI455X (gfx1250) — compile-verified
//
#include <hip/hip_runtime.h>

typedef __attribute__((ext_vector_type(2))) float v2f;
typedef __attribute__((ext_vector_type(8))) float v8f;

#define Bn 2
#define Ln 1024
#define Cn 32
#define Rn 16
#define Tn 32
#define NCHUNK (Ln / Tn)   /* 32  */
#define NSEQ   (Bn * Cn)   /* 64  */

// ---------------------------------------------------------------------------
// Kernel 1: per-chunk reduction.
//   For chunk k of sequence (b,c):
//     P      = A[t0+T-1] @ ... @ A[t0]        (complex 16x16, via WMMA)
//     y_end  = local recurrence with zero carry (complex 16-vec, via shuffles)
//   One wave (32 lanes) per chunk; 2048 waves total.
// ---------------------------------------------------------------------------
__global__ __launch_bounds__(32)
void pscan_chunk_reduce(const float* __restrict__ A_r, const float* __restrict__ A_i,
                        const float* __restrict__ X_r, const float* __restrict__ X_i,
                        float* __restrict__ wsP, float* __restrict__ wsYend) {
    __shared__ __align__(16) float sPr[256];
    __shared__ __align__(16) float sPi[256];

    const int lane = threadIdx.x;
    const int hi   = lane >> 4;      // 0 for lanes 0-15, 1 for lanes 16-31
    const int m    = lane & 15;      // row (A-operand) / col (B,D operands)
    const int blk  = blockIdx.x;
    const int seq  = blk / NCHUNK;
    const int k    = blk % NCHUNK;
    const int b    = seq / Cn;
    const int c    = seq % Cn;
    const int t0   = k * Tn;

    // P := A[t0] into LDS (row-major), b128 copies
    {
        const size_t a0 = (size_t)((b * Ln + t0) * Cn + c) * (Rn * Rn);
        const float4* ar0 = (const float4*)(A_r + a0);
        const float4* ai0 = (const float4*)(A_i + a0);
        ((float4*)sPr)[lane]      = ar0[lane];
        ((float4*)sPr)[lane + 32] = ar0[lane + 32];
        ((float4*)sPi)[lane]      = ai0[lane];
        ((float4*)sPi)[lane + 32] = ai0[lane + 32];
    }
    // y := X[t0]  (lane l holds component l&15; both halves hold copies)
    const size_t x0 = (size_t)((b * Ln + t0) * Cn + c) * Rn;
    float yr = X_r[x0 + m];
    float yi = X_i[x0 + m];

    // A-operand loader: 16x4 slices, lane = row, half-wave picks the col pair
    auto loadA = [&](int t, v2f* ar, v2f* ai) {
        const size_t abase = (size_t)((b * Ln + t) * Cn + c) * (Rn * Rn);
        const float* arp = A_r + abase + m * 16 + hi * 2;
        const float* aip = A_i + abase + m * 16 + hi * 2;
#pragma unroll
        for (int kk = 0; kk < 4; ++kk) {
            ar[kk].x = arp[kk * 4];  ar[kk].y = arp[kk * 4 + 1];
            ai[kk].x = aip[kk * 4];  ai[kk].y = aip[kk * 4 + 1];
        }
    };

    // One recurrence step using preloaded A operands for time t.
    auto step = [&](int t, const v2f* ar, const v2f* ai) {
        v2f nai[4];
#pragma unroll
        for (int kk = 0; kk < 4; ++kk) nai[kk] = -ai[kk];

        // B operands: rows 4kk..4kk+3 of P from LDS
        v2f br[4], bi[4];
#pragma unroll
        for (int kk = 0; kk < 4; ++kk) {
            const int r0 = 4 * kk + hi;
            br[kk].x = sPr[r0 * 16 + m];
            br[kk].y = sPr[(r0 + 2) * 16 + m];
            bi[kk].x = sPi[r0 * 16 + m];
            bi[kk].y = sPi[(r0 + 2) * 16 + m];
        }

        // Complex matmat: Dr = Ar@Pr - Ai@Pi ; Di = Ar@Pi + Ai@Pr
        // Four INDEPENDENT 4-deep WMMA chains (critical path = 4 WMMAs + 1 add)
        v8f dr0 = {}, dr1 = {}, di0 = {}, di1 = {};
#pragma unroll
        for (int kk = 0; kk < 4; ++kk) {
            dr0 = __builtin_amdgcn_wmma_f32_16x16x4_f32(false, ar[kk],  false, br[kk], (short)0, dr0, false, false);
            dr1 = __builtin_amdgcn_wmma_f32_16x16x4_f32(false, nai[kk], false, bi[kk], (short)0, dr1, false, false);
            di0 = __builtin_amdgcn_wmma_f32_16x16x4_f32(false, ar[kk],  false, bi[kk], (short)0, di0, false, false);
            di1 = __builtin_amdgcn_wmma_f32_16x16x4_f32(false, ai[kk],  false, br[kk], (short)0, di1, false, false);
        }

        // y = A[t] @ y + X[t] using the A-operand regs + lane shuffles
        const size_t xbase = (size_t)((b * Ln + t) * Cn + c) * Rn;
        float sr = 0.f, si = 0.f;
#pragma unroll
        for (int kk = 0; kk < 4; ++kk) {
            const int c0 = kk * 4 + hi * 2;
            float y0r = __shfl(yr, c0,     32), y0i = __shfl(yi, c0,     32);
            float y1r = __shfl(yr, c0 + 1, 32), y1i = __shfl(yi, c0 + 1, 32);
            sr += ar[kk].x * y0r - ai[kk].x * y0i;
            si += ar[kk].x * y0i + ai[kk].x * y0r;
            sr += ar[kk].y * y1r - ai[kk].y * y1i;
            si += ar[kk].y * y1i + ai[kk].y * y1r;
        }
        sr += __shfl_xor(sr, 16, 32);
        si += __shfl_xor(si, 16, 32);
        yr = sr + X_r[xbase + m];
        yi = si + X_i[xbase + m];

        // P <- D: combine split accumulators while writing back to LDS
#pragma unroll
        for (int r = 0; r < 8; ++r) {
            sPr[(hi * 8 + r) * 16 + m] = dr0[r] + dr1[r];
            sPi[(hi * 8 + r) * 16 + m] = di0[r] + di1[r];
        }
    };

    // Double-buffered main loop: next step's A operands load while current
    // step's WMMA chain runs (A loads never sit on the P critical path).
    v2f arC[4], aiC[4], arN[4], aiN[4];
    loadA(t0 + 1, arC, aiC);
    for (int t = t0 + 1; t < t0 + Tn; ++t) {
        const bool more = (t + 1 < t0 + Tn);
        if (more) loadA(t + 1, arN, aiN);
        step(t, arC, aiC);
        if (more) {
#pragma unroll
            for (int kk = 0; kk < 4; ++kk) { arC[kk] = arN[kk]; aiC[kk] = aiN[kk]; }
        }
    }

    // Write chunk summary: P (b128) and y_end
    float4* Pout = (float4*)(wsP + (size_t)(seq * NCHUNK + k) * 512);
    Pout[lane]      = ((const float4*)sPr)[lane];
    Pout[lane + 32] = ((const float4*)sPr)[lane + 32];
    Pout[lane + 64] = ((const float4*)sPi)[lane];
    Pout[lane + 96] = ((const float4*)sPi)[lane + 32];
    float* Yout = wsYend + (size_t)(seq * NCHUNK + k) * 32;
    if (lane < 16) { Yout[m] = yr; Yout[16 + m] = yi; }
}

// ---------------------------------------------------------------------------
// Kernel 2: sequential scan over the 32 chunk summaries of each sequence.
//   carry[0] = 0 ; carry[k+1] = P_k @ carry[k] + yend_k
//   Writes carry-IN per chunk. One wave per sequence.
// ---------------------------------------------------------------------------
__global__ __launch_bounds__(32)
void pscan_chunk_scan(const float* __restrict__ wsP, const float* __restrict__ wsYend,
                      float* __restrict__ wsCarry) {
    const int lane = threadIdx.x;
    const int m    = lane & 15;
    const int seq  = blockIdx.x;

    float cr = 0.f, ci = 0.f;   // lane l holds carry[l&15]
    for (int k = 0; k < NCHUNK; ++k) {
        float* Cout = wsCarry + (size_t)(seq * NCHUNK + k) * 32;
        if (lane < 16) { Cout[m] = cr; Cout[16 + m] = ci; }

        const float* P  = wsP    + (size_t)(seq * NCHUNK + k) * 512;
        const float* Ye = wsYend + (size_t)(seq * NCHUNK + k) * 32;
        float sr = 0.f, si = 0.f;
#pragma unroll 4
        for (int j = 0; j < 16; ++j) {
            const float pr = P[m * 16 + j];
            const float pi = P[256 + m * 16 + j];
            const float vr = __shfl(cr, j, 32);
            const float vi = __shfl(ci, j, 32);
            sr += pr * vr - pi * vi;
            si += pr * vi + pi * vr;
        }
        cr = sr + Ye[m];
        ci = si + Ye[16 + m];
    }
}

// ---------------------------------------------------------------------------
// Kernel 3: re-run the recurrence per chunk from its carry-in; emit Y.
//   Second read of A should be largely L2-resident (134 MB < 192 MB L2).
//   Output layout matches jnp.stack((real, imag), -1): [B,L,C,R,2].
// ---------------------------------------------------------------------------
__global__ __launch_bounds__(32)
void pscan_chunk_apply(const float* __restrict__ A_r, const float* __restrict__ A_i,
                       const float* __restrict__ X_r, const float* __restrict__ X_i,
                       const float* __restrict__ wsCarry, float* __restrict__ out) {
    const int lane = threadIdx.x;
    const int hi   = lane >> 4;
    const int m    = lane & 15;
    const int blk  = blockIdx.x;
    const int seq  = blk / NCHUNK;
    const int k    = blk % NCHUNK;
    const int b    = seq / Cn;
    const int c    = seq % Cn;
    const int t0   = k * Tn;

    const float* Cin = wsCarry + (size_t)(seq * NCHUNK + k) * 32;
    float yr = Cin[m];
    float yi = Cin[16 + m];

    for (int t = t0; t < t0 + Tn; ++t) {
        const size_t abase = (size_t)((b * Ln + t) * Cn + c) * (Rn * Rn);
        const size_t xbase = (size_t)((b * Ln + t) * Cn + c) * Rn;

        // lane handles row m, cols hi*8 .. hi*8+7 (two b128 loads per part)
        const float4* arp = (const float4*)(A_r + abase + m * 16 + hi * 8);
        const float4* aip = (const float4*)(A_i + abase + m * 16 + hi * 8);
        const float4 r0 = arp[0], r1 = arp[1];
        const float4 i0 = aip[0], i1 = aip[1];
        const float arr[8] = { r0.x, r0.y, r0.z, r0.w, r1.x, r1.y, r1.z, r1.w };
        const float aii[8] = { i0.x, i0.y, i0.z, i0.w, i1.x, i1.y, i1.z, i1.w };

        float sr = 0.f, si = 0.f;
#pragma unroll
        for (int jj = 0; jj < 8; ++jj) {
            const int j = hi * 8 + jj;
            const float vr = __shfl(yr, j, 32);
            const float vi = __shfl(yi, j, 32);
            sr += arr[jj] * vr - aii[jj] * vi;
            si += arr[jj] * vi + aii[jj] * vr;
        }
        sr += __shfl_xor(sr, 16, 32);
        si += __shfl_xor(si, 16, 32);
        yr = sr + X_r[xbase + m];
        yi = si + X_i[xbase + m];

        if (lane < 16) {   // coalesced 128B store: (re, im) pairs for 16 rows
            float2* o = (float2*)(out + (xbase + m) * 2);
            *o = make_float2(yr, yi);
        }
    }
}

// ---------------------------------------------------------------------------
extern "C" void kernel_launch(void* const* d_in, const int* in_sizes, int n_in,
                              void* d_out, int out_size, void* d_ws, size_t ws_size,
                              hipStream_t stream) {
    const float* A_r = (const float*)d_in[0];
    const float* A_i = (const float*)d_in[1];
    const float* X_r = (const float*)d_in[2];
    const float* X_i = (const float*)d_in[3];
    float* out = (float*)d_out;

    float* ws      = (float*)d_ws;
    float* wsP     = ws;                                          // 64*32*512 floats (4 MB)
    float* wsYend  = wsP    + (size_t)NSEQ * NCHUNK * 512;        // 64*32*32
    float* wsCarry = wsYend + (size_t)NSEQ * NCHUNK * 32;         // 64*32*32

    dim3 blk(32);
    pscan_chunk_reduce<<<NSEQ * NCHUNK, blk, 0, stream>>>(A_r, A_i, X_r, X_i, wsP, wsYend);
    pscan_chunk_scan  <<<NSEQ,          blk, 0, stream>>>(wsP, wsYend, wsCarry);
    pscan_chunk_apply <<<NSEQ * NCHUNK, blk, 0, stream>>>(A_r, A_i, X_r, X_i, wsCarry, out);
}